// LePEAttention_6880537608743
// MI455X (gfx1250) — compile-verified
//
#include <hip/hip_runtime.h>

// CDNA5 / gfx1250, wave32. Linear (softmax-free) window attention:
//   out = Q (K^T V) * (2*SCALE),  per (t, b, win, head) problem.
// 128 problems, each N=1568 x d=32, fp32 throughout (V_WMMA_F32_16X16X4_F32).
// Reassociation: (Q K^T) V == Q (K^T V); K^T V is only 32x32, so the whole
// computation is HBM-bandwidth-bound (~103 MB ~= 4.4 us at 23.3 TB/s).

typedef __attribute__((ext_vector_type(2))) float v2f;
typedef __attribute__((ext_vector_type(8))) float v8f;

#define T_    4
#define B_    2
#define C_    256
#define H_    56
#define W_    56
#define HEADS 8
#define DH    32          // head dim
#define WSP   28          // w_sp (window width), h_sp = 56
#define HW    (H_ * W_)   // 3136, stride between channels
#define N_    1568        // 56 * 28 tokens per window
#define NPROB 128         // 4t * 2b * 2win * 8heads
#define CHUNKS 4          // n-dimension split for stage 1
#define SCALE2 0.35355339059327373f   // 2 * (256/8)^-0.5 = 1/(2*sqrt(2))

static __device__ __forceinline__ v8f wmma_f32(v2f a, v2f b, v8f c) {
    return __builtin_amdgcn_wmma_f32_16x16x4_f32(
        false, a, false, b, (short)0, c, false, false);
}

// ---------------------------------------------------------------------------
// Stage 1: partial M = K^T V over a 392-token slice (14 h-rows of 28), WMMA.
//   D[m=dd, col=ee] += A[m=dd, k=n] * B[k=n, col=ee], A=K^T, B=V from LDS.
// Each wave holds four 16x16 quadrant accumulators = full 32x32; the 8
// waves' partials are reduced through LDS. grid (CHUNKS, 128), block 256.
// ---------------------------------------------------------------------------
__global__ __launch_bounds__(256) void kv_reduce(const float* __restrict__ k,
                                                 const float* __restrict__ v,
                                                 float* __restrict__ part) {
    __shared__ float kt[56 * 33];
    __shared__ float vt[56 * 33];
    __shared__ float red[8 * DH * DH];     // per-wave 32x32 partials
    const int chunk = blockIdx.x;          // 0..3
    const int p     = blockIdx.y;          // 0..127
    const int head  = p & 7;
    const int win   = (p >> 3) & 1;
    const int tb    = p >> 4;              // t*2 + b
    const int tid   = threadIdx.x;
    const int wave  = tid >> 5;
    const int lane  = tid & 31;
    const int lm    = lane & 15;
    const int lh    = lane >> 4;

    const long base = (long)(tb * C_ + head * DH) * HW + win * WSP;
    const float* kb = k + base;
    const float* vb = v + base;

    v8f acc00 = {}, acc01 = {}, acc10 = {}, acc11 = {};  // [ddh][eeh]

    const int hbase = chunk * 14;
    for (int sc = 0; sc < 7; ++sc) {       // 7 sub-tiles of 2 h-rows (56 n)
        const int h0 = hbase + sc * 2;
        // load 56 x 32 of K and V, dd-major so consecutive tids walk w
        for (int f = tid; f < 56 * 32; f += 256) {
            const int d   = f / 56;
            const int nn  = f - d * 56;
            const int row = (nn >= 28) ? 1 : 0;
            const int ww  = nn - row * 28;
            const long off = (long)d * HW + (long)(h0 + row) * W_ + ww;
            kt[nn * 33 + d] = kb[off];
            vt[nn * 33 + d] = vb[off];
        }
        __syncthreads();
        // 14 k-groups of 4 tokens; wave w takes groups w, w+8 (wave-uniform)
        for (int g = wave; g < 14; g += 8) {
            const int nb = g * 4 + 2 * lh;     // A/B lane-half k split
            v2f a0, a1, b0, b1;
            a0.x = kt[nb * 33 + lm];        a0.y = kt[(nb + 1) * 33 + lm];
            a1.x = kt[nb * 33 + 16 + lm];   a1.y = kt[(nb + 1) * 33 + 16 + lm];
            b0.x = vt[nb * 33 + lm];        b0.y = vt[(nb + 1) * 33 + lm];
            b1.x = vt[nb * 33 + 16 + lm];   b1.y = vt[(nb + 1) * 33 + 16 + lm];
            acc00 = wmma_f32(a0, b0, acc00);
            acc01 = wmma_f32(a0, b1, acc01);
            acc10 = wmma_f32(a1, b0, acc10);
            acc11 = wmma_f32(a1, b1, acc11);
        }
        __syncthreads();
    }

    // Spill each wave's 32x32 partial: D VGPR r -> row m = r + 8*lh, col = lm.
    #pragma unroll
    for (int r = 0; r < 8; ++r) {
        const int m = r + 8 * lh;
        red[wave * 1024 + (m)      * DH + lm]      = acc00[r];
        red[wave * 1024 + (m)      * DH + 16 + lm] = acc01[r];
        red[wave * 1024 + (m + 16) * DH + lm]      = acc10[r];
        red[wave * 1024 + (m + 16) * DH + 16 + lm] = acc11[r];
    }
    __syncthreads();

    float* o = part + ((long)chunk * NPROB + p) * (DH * DH);
    for (int e = tid; e < DH * DH; e += 256) {
        float s = 0.f;
        #pragma unroll
        for (int w = 0; w < 8; ++w) s += red[w * 1024 + e];
        o[e] = s;
    }
}

// ---------------------------------------------------------------------------
// Stage 2: M_final = (sum of chunk partials) * 2*SCALE.  grid(128), block 256.
// ---------------------------------------------------------------------------
__global__ __launch_bounds__(256) void m_finalize(const float* __restrict__ part,
                                                  float* __restrict__ mfin) {
    const int p = blockIdx.x;
    for (int e = threadIdx.x; e < DH * DH; e += 256) {
        float s = 0.f;
        #pragma unroll
        for (int c = 0; c < CHUNKS; ++c)
            s += part[((long)c * NPROB + p) * (DH * DH) + e];
        mfin[(long)p * (DH * DH) + e] = s * SCALE2;
    }
}

// ---------------------------------------------------------------------------
// Stage 3: out^T = M^T * Q^T via V_WMMA_F32_16X16X4_F32.
//   A (16x4) = M^T tile:  m = ee (from LDS),  k = dd
//   B (4x16) = Q^T tile:  k = dd, col = n  -> lane 0-15 = consecutive n (coalesced)
//   D (16x16): m = ee (VGPR index), col = n (lane)  -> coalesced stores
// grid (49, 128), block 64 = 2 waves; wave w owns 16-token tile blockIdx.x*2+w.
// 98 tiles * 16 = 1568 exactly; 16 WMMAs per wave.
// ---------------------------------------------------------------------------
__global__ __launch_bounds__(64) void qm_gemm(const float* __restrict__ q,
                                              const float* __restrict__ mfin,
                                              float* __restrict__ out) {
    __shared__ float Ms[DH * DH];          // M[dd][ee]
    const int p    = blockIdx.y;
    const int head = p & 7;
    const int win  = (p >> 3) & 1;
    const int tb   = p >> 4;
    const int tid  = threadIdx.x;

    for (int e = tid; e < DH * DH; e += 64) Ms[e] = mfin[(long)p * (DH * DH) + e];
    __syncthreads();

    const int wave = tid >> 5;
    const int lane = tid & 31;
    const int lm   = lane & 15;            // n-col / ee within half
    const int lh   = lane >> 4;            // k-pair selector (A/B lane halves)

    const int tile = blockIdx.x * 2 + wave;      // 0..97
    const int n    = tile * 16 + lm;             // token index, < 1568
    const int h    = n / 28;
    const int ww   = n - h * 28;
    const long base = (long)(tb * C_ + head * DH) * HW + win * WSP
                    + (long)h * W_ + ww;
    const float* qb = q + base;
    float* ob       = out + base;

    // B fragments: Q^T. Lane (lm,lh): k = 4s + 2*lh + {0,1}, col = n.
    v2f bfrag[8];
    #pragma unroll
    for (int s = 0; s < 8; ++s) {
        const int dd = 4 * s + 2 * lh;
        bfrag[s].x = qb[(long)dd * HW];
        bfrag[s].y = qb[(long)(dd + 1) * HW];
    }

    v8f acc0 = {};   // ee 0..15
    v8f acc1 = {};   // ee 16..31
    #pragma unroll
    for (int s = 0; s < 8; ++s) {
        const int dd = 4 * s + 2 * lh;
        v2f a0, a1;
        a0.x = Ms[dd * DH + lm];            // A[m=ee, k=dd]   = M[dd][ee]
        a0.y = Ms[(dd + 1) * DH + lm];
        a1.x = Ms[dd * DH + 16 + lm];
        a1.y = Ms[(dd + 1) * DH + 16 + lm];
        acc0 = wmma_f32(a0, bfrag[s], acc0);
        acc1 = wmma_f32(a1, bfrag[s], acc1);
    }

    // D: VGPR r holds row m = r + 8*lh, column = lm (-> this lane's n).
    #pragma unroll
    for (int r = 0; r < 8; ++r) {
        const int ee = r + 8 * lh;
        ob[(long)ee * HW]        = acc0[r];
        ob[(long)(ee + 16) * HW] = acc1[r];
    }
}

// ---------------------------------------------------------------------------
extern "C" void kernel_launch(void* const* d_in, const int* in_sizes, int n_in,
                              void* d_out, int out_size, void* d_ws, size_t ws_size,
                              hipStream_t stream) {
    const float* q = (const float*)d_in[0];
    const float* k = (const float*)d_in[1];
    const float* v = (const float*)d_in[2];
    // d_in[3] = v_lamda == 1 (output channels == C), folded into constants.
    float* out  = (float*)d_out;
    float* part = (float*)d_ws;                          // 4*128*1024 f32 = 2 MB
    float* mfin = part + (long)CHUNKS * NPROB * DH * DH; // 128*1024 f32 = 512 KB

    kv_reduce <<<dim3(CHUNKS, NPROB), 256, 0, stream>>>(k, v, part);
    m_finalize<<<dim3(NPROB),        256, 0, stream>>>(part, mfin);
    qm_gemm   <<<dim3(49, NPROB),     64, 0, stream>>>(q, mfin, out);
}